// ModelDisigner_74096775790702
// MI455X (gfx1250) — compile-verified
//
#include <hip/hip_runtime.h>

// Depthwise cross-correlation (T=16,B=16,C=256, 31x31 vs 15x15 -> 17x17)
// via Toeplitz-GEMM on V_WMMA_F32_16X16X4_F32 (exact fp32).
// Tile A: D[t,ox]  = sum_{ky,x} S[t,oy+ky,x]   * K[ky,x-ox]   (per oy, ox 0..15)
// Tile B: D[t,oy]  = sum_{y,kx} S[t,y,16+kx]   * K[y-oy,kx]   (ox=16 col, oy 0..15)
// Corner (oy=16,ox=16): VALU dot + cross-half shuffle reduce.

#define T_   16
#define B_   16
#define C_   256
#define HS_  31
#define HT_  15
#define OUT_ 17
#define OA_  289              // 17*17
#define CORR_ELEMS (B_ * T_ * C_ * OA_)   // 18,939,904

typedef __attribute__((ext_vector_type(2))) float v2f;
typedef __attribute__((ext_vector_type(8))) float v8f;

// LDS geometry: all pitches even (8B-aligned b64) and bank-conflict free
// (pitch mod 64 has gcd 4 with 64 -> 16 distinct bank groups, multiples of 4)
#define XPITCH  36                    // padded x pitch for S rows (31 -> 36)
#define TPITCH  (HS_ * XPITCH)        // 1116 dwords per search image
#define S_DW    (T_ * TPITCH)         // 17856 dwords
#define KDIM    480                   // tile-A contraction: 15 ky * 32 x
#define BPITCH  484
#define B_DW    (16 * BPITCH)         // 7744 dwords
#define KD2     480                   // tile-B contraction: 30 y * 16 kx
#define B2PITCH 500
#define B2_DW   (16 * B2PITCH)        // 8000 dwords
#define K_DW    240                   // template 225, padded
#define SMEM_DW (S_DW + B_DW + B2_DW + K_DW)  // 33840 dwords = ~135 KB

#define WMMA_F32(a, b, c) \
  __builtin_amdgcn_wmma_f32_16x16x4_f32(false, (a), false, (b), (short)0, (c), false, false)

__global__ __launch_bounds__(128)
void dwcorr_wmma(const float* __restrict__ s_f,
                 const float* __restrict__ t_f,
                 float* __restrict__ corr) {
  extern __shared__ float smem[];
  float* sS  = smem;                         // [t][y][x]   pitch XPITCH/TPITCH
  float* sB  = smem + S_DW;                  // [ox][k]     k=(ky,x)  pitch BPITCH
  float* sB2 = smem + S_DW + B_DW;           // [oy][k]     k=(y,kx)  pitch B2PITCH
  float* sK  = smem + S_DW + B_DW + B2_DW;   // [225] template

  const int bc  = blockIdx.x;                // b*256 + c
  const int b   = bc >> 8;
  const int c   = bc & 255;
  const int tid = threadIdx.x;

  // ---- phase 0: zero LDS (pads must be 0) -------------------------------
  for (int i = tid; i < SMEM_DW; i += 128) smem[i] = 0.0f;
  __syncthreads();

  // ---- phase 1: stage search images + template --------------------------
  for (int j = tid; j < T_ * HS_ * HS_; j += 128) {
    int t = j / (HS_ * HS_);
    int e = j - t * (HS_ * HS_);
    int y = e / HS_;
    int x = e - y * HS_;
    sS[t * TPITCH + y * XPITCH + x] =
        s_f[(((size_t)t * B_ + b) * C_ + c) * (HS_ * HS_) + e];
  }
  for (int j = tid; j < HT_ * HT_; j += 128)
    sK[j] = t_f[((size_t)b * C_ + c) * (HT_ * HT_) + j];
  __syncthreads();

  // ---- phase 2: build both Toeplitz operands ----------------------------
  for (int j = tid; j < 16 * KDIM; j += 128) {     // sB[(ky,x), ox]=K[ky,x-ox]
    int n = j / KDIM;
    int k = j - n * KDIM;
    int x  = k & 31;
    int ky = k >> 5;
    int tap = x - n;
    float v = 0.0f;
    if (x <= 30 && tap >= 0 && tap < HT_) v = sK[ky * HT_ + tap];
    sB[n * BPITCH + k] = v;
  }
  for (int j = tid; j < 16 * KD2; j += 128) {      // sB2[(y,kx), oy]=K[y-oy,kx]
    int n = j / KD2;
    int k = j - n * KD2;
    int kx = k & 15;
    int y  = k >> 4;
    int tap = y - n;
    float v = 0.0f;
    if (kx < HT_ && tap >= 0 && tap < HT_) v = sK[tap * HT_ + kx];
    sB2[n * B2PITCH + k] = v;
  }
  __syncthreads();

  // ---- phase 3: GEMMs ----------------------------------------------------
  const int lane = tid & 31;
  const int wave = tid >> 5;          // 4 waves
  const int hi   = lane >> 4;         // 0: K0/K1 half, 1: K2/K3 half
  const int nl   = lane & 15;

  const float* aLane = sS + nl * TPITCH + 2 * hi;  // A: M = t = nl
  const float* bLane = sB + nl * BPITCH + 2 * hi;  // B: N = ox = nl
  const int bcBase = (b * T_ * C_ + c) * OA_;      // + t*(C_*OA_) per row

  // Tile A: per oy, ox = 0..15. Two accumulator chains by ky parity.
  for (int oy = wave; oy < OUT_; oy += 4) {
    v8f accA = {};
    v8f accB = {};
    const float* aRow = aLane + oy * XPITCH;
    for (int kp = 0; kp < 7; ++kp) {               // ky = 2kp, 2kp+1
      const float* aA = aRow  + (2 * kp) * XPITCH;
      const float* aB = aA + XPITCH;
      const float* bA = bLane + (2 * kp) * 32;
      const float* bB = bA + 32;
#pragma unroll
      for (int xs = 0; xs < 8; ++xs) {             // K = 32 per row, 4 at a time
        v2f a0 = *(const v2f*)(aA + xs * 4);
        v2f b0 = *(const v2f*)(bA + xs * 4);
        accA = WMMA_F32(a0, b0, accA);
        v2f a1 = *(const v2f*)(aB + xs * 4);
        v2f b1 = *(const v2f*)(bB + xs * 4);
        accB = WMMA_F32(a1, b1, accB);
      }
    }
    {                                              // ky = 14 -> chain A
      const float* aA = aRow  + 14 * XPITCH;
      const float* bA = bLane + 14 * 32;
#pragma unroll
      for (int xs = 0; xs < 8; ++xs) {
        v2f a0 = *(const v2f*)(aA + xs * 4);
        v2f b0 = *(const v2f*)(bA + xs * 4);
        accA = WMMA_F32(a0, b0, accA);
      }
    }
    v8f acc = accA + accB;
    // D: lane(0-15) N=lane, VGPR r -> M=r ; lanes 16-31 -> M=8+r
#pragma unroll
    for (int r = 0; r < 8; ++r) {
      int t = r + 8 * hi;
      corr[bcBase + t * (C_ * OA_) + oy * OUT_ + nl] = acc[r];
    }
  }

  if (wave == 3) {
    // Tile B: ox = 16 column for oy = 0..15 in one GEMM (N = oy).
    const float* a2 = sS  + nl * TPITCH + 16 + 2 * hi;
    const float* b2 = sB2 + nl * B2PITCH + 2 * hi;
    v8f accC = {};
    v8f accD = {};
    for (int yp = 0; yp < 15; ++yp) {              // y = 2yp, 2yp+1 (0..29)
      const float* aC = a2 + (2 * yp) * XPITCH;
      const float* aD = aC + XPITCH;
      const float* bC = b2 + (2 * yp) * 16;
      const float* bD = bC + 16;
#pragma unroll
      for (int xs = 0; xs < 4; ++xs) {             // K = 16 per row
        v2f a0 = *(const v2f*)(aC + xs * 4);
        v2f b0 = *(const v2f*)(bC + xs * 4);
        accC = WMMA_F32(a0, b0, accC);
        v2f a1 = *(const v2f*)(aD + xs * 4);
        v2f b1 = *(const v2f*)(bD + xs * 4);
        accD = WMMA_F32(a1, b1, accD);
      }
    }
    v8f accE = accC + accD;
#pragma unroll
    for (int r = 0; r < 8; ++r) {                  // lane nl holds column oy=nl
      int t = r + 8 * hi;
      corr[bcBase + t * (C_ * OA_) + nl * OUT_ + 16] = accE[r];
    }

    // Corner element (oy=16, ox=16): 225-MAC dot per t, split across halves.
    {
      int t = nl;
      float v = 0.0f;
      int ky0 = hi * 8;
      int kyN = hi ? 7 : 8;                        // ky 0..7 | 8..14
      for (int i = 0; i < kyN; ++i) {
        int ky = ky0 + i;
        const float* sr = sS + t * TPITCH + (16 + ky) * XPITCH + 16;
        const float* kr = sK + ky * HT_;
#pragma unroll
        for (int kx = 0; kx < HT_; ++kx) v += sr[kx] * kr[kx];
      }
      v += __shfl_xor(v, 16, 32);
      if (hi == 0)
        corr[bcBase + t * (C_ * OA_) + 16 * OUT_ + 16] = v;
    }
  }
}

// masks_feat[b][t][c] = corr[b][t][c][p0][p1]
__global__ __launch_bounds__(256)
void gather_masks(const float* __restrict__ corr,
                  const int* __restrict__ pos,
                  float* __restrict__ mout) {
  int i = blockIdx.x * 256 + threadIdx.x;      // 65536 total
  int b = i >> 12;
  int t = (i >> 8) & 15;
  int c = i & 255;
  int p0 = pos[(b * T_ + t) * 2 + 0];
  int p1 = pos[(b * T_ + t) * 2 + 1];
  mout[i] = corr[((b * T_ + t) * C_ + c) * OA_ + p0 * OUT_ + p1];
}

extern "C" void kernel_launch(void* const* d_in, const int* in_sizes, int n_in,
                              void* d_out, int out_size, void* d_ws, size_t ws_size,
                              hipStream_t stream) {
  (void)in_sizes; (void)n_in; (void)d_ws; (void)ws_size; (void)out_size;
  const float* s_f = (const float*)d_in[0];
  const float* t_f = (const float*)d_in[1];
  const int*   pos = (const int*)d_in[2];
  float* corr  = (float*)d_out;
  float* masks = corr + CORR_ELEMS;

  size_t shbytes = (size_t)SMEM_DW * sizeof(float);
  dwcorr_wmma<<<dim3(B_ * C_), dim3(128), shbytes, stream>>>(s_f, t_f, corr);
  gather_masks<<<dim3((B_ * T_ * C_) / 256), dim3(256), 0, stream>>>(corr, pos, masks);
}